// PerIMUKinematicsGenerator_16587163697395
// MI455X (gfx1250) — compile-verified
//
#include <hip/hip_runtime.h>

typedef float v2f __attribute__((ext_vector_type(2)));
typedef float v8f __attribute__((ext_vector_type(8)));

#define T_STEPS   256
#define TILE_P    128          // outputs per block (8 waves x 16)
#define NSLOT     384          // staged s-window: [P-256, P+128)
#define NTHREADS  256

__global__ __launch_bounds__(NTHREADS)
void kin_band_wmma_kernel(const float* __restrict__ k1,  const float* __restrict__ d1,
                          const float* __restrict__ ph1, const float* __restrict__ c1,
                          const float* __restrict__ k2,  const float* __restrict__ d2,
                          const float* __restrict__ ph2, const float* __restrict__ c2,
                          float* __restrict__ out, int S)
{
    // Derived params, AoS float4 per oscillator: {c, d/2, omega, phi}
    __shared__ float4 pA[NSLOT];
    __shared__ float4 pB[NSLOT];
    __shared__ float  dtile[8 * 256];   // per-wave 16x16 D-tile scratch

    const int tid = threadIdx.x;
    const int P   = blockIdx.x * TILE_P;

    // ---- Stage sliding window + hoist per-s transcendentals (sqrt) ----
    for (int i = tid; i < NSLOT; i += NTHREADS) {
        int s = P - 256 + i;
        float4 a = {0.0f, 0.0f, 0.0f, 0.0f};
        float4 b = {0.0f, 0.0f, 0.0f, 0.0f};
        if (s >= 0 && s < S) {
            float kk = k1[s], dd = d1[s];
            a.x = c1[s];
            a.y = 0.5f * dd;
            a.z = 0.5f * __builtin_sqrtf(4.0f * kk - dd * dd);
            a.w = ph1[s];
            kk = k2[s]; dd = d2[s];
            b.x = c2[s];
            b.y = 0.5f * dd;
            b.z = 0.5f * __builtin_sqrtf(4.0f * kk - dd * dd);
            b.w = ph2[s];
        }
        pA[i] = a;
        pB[i] = b;
    }
    __syncthreads();

    // ---- Per-wave 16-output tile; accumulate via V_WMMA_F32_16X16X4_F32 ----
    const int wave = tid >> 5;
    const int lane = tid & 31;
    const int m    = lane & 15;            // output row within tile
    const int kb   = (lane >> 4) << 1;     // K base: 0 (lanes 0-15) / 2 (lanes 16-31)

    // slot index for (p = P + wave*16 + m, j): idx = (p - P) + 256 - j
    int   idx = wave * 16 + m + 256 - kb;  // j = kb at chunk 0
    float tj  = (float)kb;

    v8f acc = {};                          // C/D accumulator (16x16 f32)
    const v2f bones = {1.0f, 1.0f};        // B = all-ones: every D column = row sum

    #pragma unroll 4
    for (int ch = 0; ch < T_STEPS / 4; ++ch) {
        float4 qa0 = pA[idx];     float4 qb0 = pB[idx];
        float4 qa1 = pA[idx - 1]; float4 qb1 = pB[idx - 1];
        float t0 = tj, t1 = tj + 1.0f;

        v2f a;
        a.x = qa0.x * __expf(-qa0.y * t0) * __sinf(fmaf(t0, qa0.z, qa0.w))
            + qb0.x * __expf(-qb0.y * t0) * __sinf(fmaf(t0, qb0.z, qb0.w));
        a.y = qa1.x * __expf(-qa1.y * t1) * __sinf(fmaf(t1, qa1.z, qa1.w))
            + qb1.x * __expf(-qb1.y * t1) * __sinf(fmaf(t1, qb1.z, qb1.w));

        // D = A(16x4 f32) x B(4x16 ones) + C  -> every column holds row sums
        acc = __builtin_amdgcn_wmma_f32_16x16x4_f32(
                  false, a, false, bones, (short)0, acc, false, false);

        idx -= 4;
        tj  += 4.0f;
    }

    // ---- Extract row sums: lane L, VGPR r holds D[M = r + 8*(L>>4), N = L&15] ----
    float* dt = dtile + wave * 256;
    const int mrow = (lane >> 4) * 8;
    const int ncol = lane & 15;
    #pragma unroll
    for (int r = 0; r < 8; ++r)
        dt[(mrow + r) * 16 + ncol] = acc[r];
    __syncthreads();

    if (lane < 16) {
        int p = P + wave * 16 + lane;
        if (p < S) out[p] = dt[lane * 16 + 0];   // N=0 column = row sum
    }
}

extern "C" void kernel_launch(void* const* d_in, const int* in_sizes, int n_in,
                              void* d_out, int out_size, void* d_ws, size_t ws_size,
                              hipStream_t stream) {
    const float* k1  = (const float*)d_in[0];
    const float* dd1 = (const float*)d_in[1];
    const float* ph1 = (const float*)d_in[2];
    const float* c1  = (const float*)d_in[3];
    const float* k2  = (const float*)d_in[4];
    const float* dd2 = (const float*)d_in[5];
    const float* ph2 = (const float*)d_in[6];
    const float* c2  = (const float*)d_in[7];
    float* out = (float*)d_out;

    const int S = in_sizes[0];                       // 131072
    const int nblocks = (S + TILE_P - 1) / TILE_P;   // 1024

    kin_band_wmma_kernel<<<nblocks, NTHREADS, 0, stream>>>(
        k1, dd1, ph1, c1, k2, dd2, ph2, c2, out, S);
}